// SO3Linear_7232724927036
// MI455X (gfx1250) — compile-verified
//
#include <hip/hip_runtime.h>
#include <math.h>

typedef float v2f __attribute__((ext_vector_type(2)));
typedef float v8f __attribute__((ext_vector_type(8)));
typedef unsigned int v4u __attribute__((ext_vector_type(4)));
typedef int v4i __attribute__((ext_vector_type(4)));
typedef int v8i __attribute__((ext_vector_type(8)));
typedef int v4i_vs __attribute__((vector_size(16)));   // gcc-style, matches builtin sig

#define G       16      // nodes per workgroup
#define CIN     128
#define COUT    128
#define IRR     16
#define NPATHS  34
#define TSTRIDE 132     // T column stride in floats (bank-conflict padding)

#if defined(__gfx1250__) && __has_builtin(__builtin_amdgcn_global_load_async_to_lds_b128)
#define HAVE_ASYNC_LDS 1
#else
#define HAVE_ASYNC_LDS 0
#endif
#if defined(__gfx1250__) && __has_builtin(__builtin_amdgcn_tensor_load_to_lds)
#define HAVE_TDM 1
#else
#define HAVE_TDM 0
#endif

// Path tables, enumerated exactly as the reference: for lo: for l1: for l2.
__constant__ int P_L1[NPATHS] = {0,1,2,3, 0,1,1,1,2,2,2,3,3, 0,1,1,1,2,2,2,2,3,3,3, 0,1,1,2,2,2,3,3,3,3};
__constant__ int P_L2[NPATHS] = {0,1,2,3, 1,0,1,2,1,2,3,2,3, 2,1,2,3,0,1,2,3,1,2,3, 3,2,3,1,2,3,0,1,2,3};
__constant__ int P_LO[NPATHS] = {0,0,0,0, 1,1,1,1,1,1,1,1,1, 2,2,2,2,2,2,2,2,2,2,2, 3,3,3,3,3,3,3,3,3,3};
__constant__ float P_SCALE[NPATHS] = {
  0.5f,0.5f,0.5f,0.5f,
  0.33333333f,0.33333333f,0.33333333f,0.33333333f,0.33333333f,0.33333333f,0.33333333f,0.33333333f,0.33333333f,
  0.30151135f,0.30151135f,0.30151135f,0.30151135f,0.30151135f,0.30151135f,0.30151135f,0.30151135f,0.30151135f,0.30151135f,0.30151135f,
  0.31622776f,0.31622776f,0.31622776f,0.31622776f,0.31622776f,0.31622776f,0.31622776f,0.31622776f,0.31622776f,0.31622776f};

// ---------------------------------------------------------------------------
// CDNA5 data movement helpers (guarded; fall back to synchronous copies)
// ---------------------------------------------------------------------------
__device__ inline void stage_copy(float* dst_lds, const float* src_g, int nfloats, int tid) {
#if HAVE_ASYNC_LDS
  for (int i = tid*4; i < nfloats; i += 256*4)
    __builtin_amdgcn_global_load_async_to_lds_b128(
        (__attribute__((address_space(1))) v4i_vs*)(src_g + i),
        (__attribute__((address_space(3))) v4i_vs*)(dst_lds + i), 0, 0);
#else
  for (int i = tid*4; i < nfloats; i += 256*4)
    *(float4*)(dst_lds + i) = *(const float4*)(src_g + i);
#endif
}

__device__ inline void wait_async_lds() {
#if HAVE_ASYNC_LDS
#if __has_builtin(__builtin_amdgcn_s_wait_asynccnt)
  __builtin_amdgcn_s_wait_asynccnt(0);
#else
  asm volatile("s_wait_asynccnt 0x0" ::: "memory");
#endif
#endif
}

#if HAVE_TDM
// 1-D contiguous global->LDS DMA via the Tensor Data Mover.
// D# built per CDNA5 ISA 8.3/8.4: data_size=3 (8B), tensor_dim0=tile_dim0=n8,
// tensor_dim1=tile_dim1=1, stride0=n8, type=2. nbytes must be a multiple of 8,
// n8 < 65536.
__device__ inline void tdm_load_contig(const float* gsrc, float* dst_lds, unsigned nbytes) {
  unsigned long long ga = (unsigned long long)(uintptr_t)gsrc;
  unsigned lds_off = (unsigned)(uintptr_t)(__attribute__((address_space(3))) float*)dst_lds;
  unsigned n8 = nbytes >> 3;
  v4u g0;
  g0.x = 1u;                                                  // count=1 (valid)
  g0.y = lds_off;                                             // lds_addr (bytes)
  g0.z = (unsigned)(ga & 0xFFFFFFFFu);                        // global_addr[31:0]
  g0.w = (unsigned)((ga >> 32) & 0x01FFFFFFu) | 0x80000000u;  // addr[56:32] | type=2
  v8i g1;
  g1[0] = (int)0x00030000u;                 // wg_mask=0, data_size=3 (8 bytes)
  g1[1] = (int)((n8 & 0xFFFFu) << 16);      // tensor_dim0[15:0] @ bit48
  g1[2] = (int)((n8 >> 16) | (1u << 16));   // tensor_dim0[31:16], tensor_dim1=1
  g1[3] = (int)(n8 << 16);                  // tile_dim0 = n8 (16-bit)
  g1[4] = 1;                                // tile_dim1=1, tile_dim2=0
  g1[5] = (int)n8;                          // tensor_dim0_stride[31:0]
  g1[6] = 0;                                // stride0 hi, stride1 lo
  g1[7] = 0;
  v4i gz = {0,0,0,0};
#if __clang_major__ >= 23
  v8i gz8 = {0,0,0,0,0,0,0,0};
  __builtin_amdgcn_tensor_load_to_lds(g0, g1, gz, gz, gz8, 0);
#else
  __builtin_amdgcn_tensor_load_to_lds(g0, g1, gz, gz, 0);
#endif
}
#endif

// ---------------------------------------------------------------------------
// CG setup: compute real-basis Clebsch-Gordan tensors into d_ws (34 x 7x7x7 f32)
// ---------------------------------------------------------------------------
struct cplx { double re, im; };
__device__ inline cplx cmul(cplx a, cplx b) {
  return {a.re*b.re - a.im*b.im, a.re*b.im + a.im*b.re};
}
__device__ inline double dfact(int n) {
  double r = 1.0;
  for (int i = 2; i <= n; ++i) r *= (double)i;
  return r;
}
__device__ double su2_cg_elem(int j1, int j2, int j3, int m1, int m2) {
  int m3 = m1 + m2;
  if (m3 < -j3 || m3 > j3) return 0.0;
  double pref = sqrt((double)(2*j3+1) * dfact(j3+j1-j2) * dfact(j3-j1+j2) * dfact(j1+j2-j3) / dfact(j1+j2+j3+1));
  pref *= sqrt(dfact(j3+m3)*dfact(j3-m3)*dfact(j1-m1)*dfact(j1+m1)*dfact(j2-m2)*dfact(j2+m2));
  double s = 0.0;
  for (int k = 0; k <= j1 + j2; ++k) {
    int d0=k, d1=j1+j2-j3-k, d2=j1-m1-k, d3=j2+m2-k, d4=j3-j2+m1+k, d5=j3-j1-m2+k;
    if (d0<0||d1<0||d2<0||d3<0||d4<0||d5<0) continue;
    double term = 1.0/(dfact(d0)*dfact(d1)*dfact(d2)*dfact(d3)*dfact(d4)*dfact(d5));
    s += (k & 1) ? -term : term;
  }
  return pref * s;
}
__device__ void fill_q(int l, cplx q[7][7]) {
  for (int i = 0; i < 7; ++i)
    for (int j = 0; j < 7; ++j) q[i][j] = {0.0, 0.0};
  double r2 = 1.0 / sqrt(2.0);
  for (int m = -l; m < 0; ++m) {
    q[l+m][l-m] = { r2, 0.0};   // l + |m|
    q[l+m][l+m] = {0.0, -r2};   // l - |m|
  }
  q[l][l] = {1.0, 0.0};
  for (int m = 1; m <= l; ++m) {
    double sgn = (m & 1) ? -1.0 : 1.0;
    q[l+m][l+m] = {sgn*r2, 0.0};
    q[l+m][l-m] = {0.0, sgn*r2};
  }
  cplx ph;
  switch (l & 3) {
    case 0: ph = { 1.0,  0.0}; break;
    case 1: ph = { 0.0, -1.0}; break;
    case 2: ph = {-1.0,  0.0}; break;
    default: ph = { 0.0,  1.0}; break;
  }
  for (int i = 0; i < 2*l+1; ++i)
    for (int j = 0; j < 2*l+1; ++j) q[i][j] = cmul(ph, q[i][j]);
}

__global__ void cg_setup_kernel(float* __restrict__ cg) {
  int p = blockIdx.x;
  int l1 = P_L1[p], l2 = P_L2[p], lo = P_LO[p];
  int n1 = 2*l1+1, n2 = 2*l2+1, n3 = 2*lo+1;
  __shared__ double C[7][7][7];
  __shared__ cplx q1[7][7], q2[7][7], q3[7][7];
  int tid = threadIdx.x;
  if (tid == 0) { fill_q(l1, q1); fill_q(l2, q2); fill_q(lo, q3); }
  for (int idx = tid; idx < 343; idx += blockDim.x) {
    int a = idx / 49, b = (idx / 7) % 7, c = idx % 7;
    double v = 0.0;
    if (a < n1 && b < n2 && c < n3) {
      int m1 = a - l1, m2 = b - l2, m3 = c - lo;
      if (m1 + m2 == m3) v = su2_cg_elem(l1, l2, lo, m1, m2);
    }
    C[a][b][c] = v;
  }
  __syncthreads();
  // W[i,j,k] = Re( sum_{a,b,c} q1[a][i] q2[b][j] q3[c][k] C[a,b,c] )
  for (int idx = tid; idx < n1*n2*n3; idx += blockDim.x) {
    int i = idx / (n2*n3), j = (idx / n3) % n2, k = idx % n3;
    cplx s = {0.0, 0.0};
    for (int a = 0; a < n1; ++a)
      for (int b = 0; b < n2; ++b)
        for (int c = 0; c < n3; ++c) {
          double cv = C[a][b][c];
          if (cv != 0.0) {
            cplx t = cmul(q1[a][i], q2[b][j]);
            t = cmul(t, q3[c][k]);
            s.re += t.re * cv; s.im += t.im * cv;
          }
        }
    cg[p*343 + i*49 + j*7 + k] = (float)s.re;
  }
}

// ---------------------------------------------------------------------------
// Main kernel: per-node CG contraction (VALU) + per-path channel GEMM (WMMA f32)
// ---------------------------------------------------------------------------
__constant__ int LO_P0[4] = {0, 4, 13, 24};
__constant__ int LO_P1[4] = {4, 13, 24, 34};

__global__ __launch_bounds__(256)
void so3_main_kernel(const float* __restrict__ x1, const float* __restrict__ x2,
                     const float* __restrict__ w,  const float* __restrict__ cg,
                     float* __restrict__ out) {
  extern __shared__ float lds[];
  float* x1s = lds;                       // [G][IRR][CIN]      32768 f
  float* x2s = x1s + G*IRR*CIN;           // [G][IRR]             256 f
  float* Ws  = x2s + G*IRR;               // [CIN][COUT]        16384 f
  float* Ts  = Ws  + CIN*COUT;            // [112][TSTRIDE]     14784 f
  float* Ss  = Ts  + 112*TSTRIDE;         // [G][49]              784 f

  const int tid   = threadIdx.x;
  const int wave  = tid >> 5;             // 0..7 : M-tile (16 output channels)
  const int lane  = tid & 31;
  const int m     = lane & 15;
  const int khalf = lane >> 4;            // 0 or 1
  const int vbase = wave * 16;
  const long node_base = (long)blockIdx.x * G;

  // Stage x1 block (16 nodes x 16 rows x 128 ch) via async-to-LDS; x2 sync.
  stage_copy(x1s, x1 + node_base * IRR * CIN, G*IRR*CIN, tid);
  for (int i = tid; i < G*IRR; i += 256)
    x2s[i] = x2[node_base*IRR + i];
  wait_async_lds();
  __syncthreads();

#pragma unroll
  for (int lo = 0; lo < 4; ++lo) {
    const int nlo    = 2*lo + 1;
    const int ntiles = nlo;               // N-tiles of 16 cols (cols = G*nlo)
    const int ncols  = G * nlo;
    v8f acc[7];
#pragma unroll
    for (int t = 0; t < ntiles; ++t) acc[t] = (v8f){0,0,0,0,0,0,0,0};

    for (int p = LO_P0[lo]; p < LO_P1[lo]; ++p) {
      const int l1 = P_L1[p], l2 = P_L2[p];
      const int n1 = 2*l1+1, n2 = 2*l2+1;
      const int off1 = l1*l1, off2 = l2*l2;
      const float scale = P_SCALE[p];
      const float* cgp = cg + p*343;

      __syncthreads();  // prior readers of Ws/Ts/Ss done

      // (1) launch W[p] staging (u-major, v contiguous) - overlapped with (2)+(3)
      const float* wg = w + (long)p * CIN * COUT;
#if HAVE_TDM
      if (wave == 0)
        tdm_load_contig(wg, Ws, CIN*COUT*4u);   // one TDM DMA per workgroup
#else
      stage_copy(Ws, wg, CIN*COUT, tid);
#endif

      // (2) per-node small matrices S[n][c*n1+a] = scale * sum_b CG[a,b,c]*x2[n,off2+b]
      for (int i = tid; i < G*nlo*n1; i += 256) {
        int n = i / (nlo*n1); int r = i % (nlo*n1);
        int c = r / n1;       int a = r % n1;
        float s = 0.f;
        for (int b = 0; b < n2; ++b)
          s += cgp[a*49 + b*7 + c] * x2s[n*IRR + off2 + b];
        Ss[n*49 + c*n1 + a] = s * scale;
      }
      __syncthreads();

      // (3) T[col][u] = sum_a S[n][c,a] * x1[n][off1+a][u],  col = n*nlo + c
      for (int i = tid; i < ncols*CIN; i += 256) {
        int col = i >> 7;  int u = i & 127;
        int n = col / nlo; int c = col % nlo;
        const float* xr = x1s + n*IRR*CIN + off1*CIN + u;
        const float* sr = Ss + n*49 + c*n1;
        float s = 0.f;
        for (int a = 0; a < n1; ++a) s += sr[a] * xr[a*CIN];
        Ts[col*TSTRIDE + u] = s;
      }
      // W staging must have landed before the WMMA phase reads Ws.
      wait_async_lds();
#if HAVE_TDM
      __builtin_amdgcn_s_wait_tensorcnt(0);
#endif
      __syncthreads();

      // (4) outT[v, col] += W_p^T x T_p^T  via V_WMMA_F32_16X16X4_F32
      //  A (16x4): lane m = M(=v), vgpr k pair at k0 = kb + 2*khalf -> W[u=k][vbase+m]
      //  B (4x16): lane m = N(=col), same k pair                     -> T[col][u=k]
      for (int kb = 0; kb < CIN; kb += 4) {
        const int k0 = kb + 2*khalf;
        v2f afrag;
        afrag.x = Ws[(k0    ) * COUT + vbase + m];
        afrag.y = Ws[(k0 + 1) * COUT + vbase + m];
#pragma unroll
        for (int t = 0; t < ntiles; ++t) {
          v2f bfrag = *(const v2f*)(Ts + (t*16 + m) * TSTRIDE + k0);
          acc[t] = __builtin_amdgcn_wmma_f32_16x16x4_f32(
              false, afrag, false, bfrag, (short)0, acc[t], false, false);
        }
      }
    } // paths

    // Epilogue: transpose through LDS (reuse Ts as [col][v]) then coalesced store.
    __syncthreads();
#pragma unroll
    for (int t = 0; t < ntiles; ++t) {
      const int col = t*16 + m;
#pragma unroll
      for (int j = 0; j < 8; ++j)          // D layout: M = j + 8*khalf, N = m
        Ts[col*TSTRIDE + vbase + j + 8*khalf] = acc[t][j];
    }
    __syncthreads();
    for (int i = tid*4; i < ncols*COUT; i += 256*4) {
      int col = i >> 7;  int v = i & 127;
      int n = col / nlo; int c = col % nlo;
      *(float4*)(out + (node_base + n)*IRR*COUT + (lo*lo + c)*COUT + v) =
          *(float4*)(Ts + col*TSTRIDE + v);
    }
  } // lo
}

// ---------------------------------------------------------------------------
extern "C" void kernel_launch(void* const* d_in, const int* in_sizes, int n_in,
                              void* d_out, int out_size, void* d_ws, size_t ws_size,
                              hipStream_t stream) {
  const float* x1 = (const float*)d_in[0];   // (16384, 16, 128) f32
  const float* x2 = (const float*)d_in[1];   // (16384, 16) f32
  const float* w  = (const float*)d_in[2];   // (34, 128, 128) f32
  float* out = (float*)d_out;                // (16384, 16, 128) f32
  float* cg  = (float*)d_ws;                 // 34 * 343 f32 CG tensors

  cg_setup_kernel<<<NPATHS, 256, 0, stream>>>(cg);

  size_t smem = (size_t)(G*IRR*CIN + G*IRR + CIN*COUT + 112*TSTRIDE + G*49) * sizeof(float);
  so3_main_kernel<<<16384 / G, 256, smem, stream>>>(x1, x2, w, cg, out);
}